// ContactSampleNet_40183714021753
// MI455X (gfx1250) — compile-verified
//
#include <hip/hip_runtime.h>
#include <cstdint>

typedef __attribute__((ext_vector_type(2))) float v2f;
typedef __attribute__((ext_vector_type(8))) float v8f;

#define NPTS 32768
#define MQ   1024
#define NB   4
#define CHUNK 512
#define TPC  (CHUNK / 16)        // tiles per chunk = 32
#define BUFSZ (TPC * 32 * 8)     // floats per staging buffer (32 KB)

// ---------------------------------------------------------------------------
// Kernel A: pack x into WMMA-B-friendly float4 (xx, xy, xz, |x|^2)
// ---------------------------------------------------------------------------
__global__ __launch_bounds__(256) void pack_x_kernel(const float* __restrict__ x,
                                                     float4* __restrict__ P) {
  int tid = blockIdx.x * 256 + threadIdx.x;          // B*N threads
  int b = tid >> 15;
  int n = tid & (NPTS - 1);
  float xx = x[(b * 3 + 0) * NPTS + n];
  float xy = x[(b * 3 + 1) * NPTS + n];
  float xz = x[(b * 3 + 2) * NPTS + n];
  P[tid] = make_float4(xx, xy, xz, xx * xx + xy * xy + xz * xz);
}

// ---------------------------------------------------------------------------
// One 16x16 output tile of y = act @ W^T via V_WMMA_F32_16X16X4_F32.
// act: LDS activations [4][stride] (rows >=4 masked to zero)
// W:   row-major [fan_out][fan_in]
// Lane layout (ISA 7.12.2): A 16x4: m = lane&15, K = {0,1} (lanes 0-15) /
// {2,3} (lanes 16-31). B 4x16: n = lane&15, same K split.
// ---------------------------------------------------------------------------
__device__ __forceinline__ v8f gemm_tile(const float* act, int stride, int fan_in,
                                         const float* __restrict__ W, int nf0,
                                         int lane) {
  int m = lane & 15;
  int arow = (m < 4) ? m : 3;
  float msk = (m < 4) ? 1.0f : 0.0f;
  int ksel = (lane < 16) ? 0 : 2;
  const float* wp = W + (size_t)(nf0 + m) * fan_in + ksel;
  const float* ap = act + arow * stride + ksel;
  v8f c = {};
  for (int k0 = 0; k0 < fan_in; k0 += 4) {
    v2f a = *(const v2f*)(ap + k0);
    a *= msk;                       // zero-pad batch rows 4..15
    v2f bb = *(const v2f*)(wp + k0);
    c = __builtin_amdgcn_wmma_f32_16x16x4_f32(false, a, false, bb, (short)0, c,
                                              false, false);
  }
  return c;
}

// ---------------------------------------------------------------------------
// Kernel B: whole MLP in one workgroup (16 waves). BN uses batch stats (4 rows).
// Emits "generated" ([B,M,3] at d_out[0..12287]) and Q float4 workspace.
// ---------------------------------------------------------------------------
__global__ __launch_bounds__(512) void mlp_q_kernel(
    const float* __restrict__ gf,
    const float* __restrict__ w1, const float* __restrict__ b1,
    const float* __restrict__ w2, const float* __restrict__ b2,
    const float* __restrict__ w3, const float* __restrict__ b3,
    const float* __restrict__ w4, const float* __restrict__ b4,
    const float* __restrict__ g1, const float* __restrict__ be1,
    const float* __restrict__ g2, const float* __restrict__ be2,
    const float* __restrict__ g3, const float* __restrict__ be3,
    float* __restrict__ gen_out, float* __restrict__ Qws) {
  __shared__ float actIn[4 * 1024];   // layer-1 input
  __shared__ float bufP[4 * 256];     // raw pre-BN output
  __shared__ float bufQ[4 * 256];     // post-BN activations
  int tid = threadIdx.x;
  int lane = tid & 31;
  int wv = tid >> 5;                  // 16 waves

  for (int i = tid; i < 4096; i += 512) actIn[i] = gf[i];
  __syncthreads();

  // ---- layer 1: [4,1024] @ [1024,256]^T ----
  {
    v8f c = gemm_tile(actIn, 1024, 1024, w1, wv * 16, lane);
    if (lane < 16) {
      int nf = wv * 16 + lane;
      #pragma unroll
      for (int v = 0; v < 4; ++v) bufP[v * 256 + nf] = c[v];
    }
  }
  __syncthreads();
  // BN1 + ReLU
  if (tid < 256) {
    float y0 = bufP[0 * 256 + tid] + b1[tid];
    float y1 = bufP[1 * 256 + tid] + b1[tid];
    float y2 = bufP[2 * 256 + tid] + b1[tid];
    float y3 = bufP[3 * 256 + tid] + b1[tid];
    float mu = 0.25f * (y0 + y1 + y2 + y3);
    float d0 = y0 - mu, d1 = y1 - mu, d2 = y2 - mu, d3 = y3 - mu;
    float var = 0.25f * (d0 * d0 + d1 * d1 + d2 * d2 + d3 * d3);
    float sc = g1[tid] * rsqrtf(var + 1e-5f);
    float sh = be1[tid];
    bufQ[0 * 256 + tid] = fmaxf(sc * d0 + sh, 0.0f);
    bufQ[1 * 256 + tid] = fmaxf(sc * d1 + sh, 0.0f);
    bufQ[2 * 256 + tid] = fmaxf(sc * d2 + sh, 0.0f);
    bufQ[3 * 256 + tid] = fmaxf(sc * d3 + sh, 0.0f);
  }
  __syncthreads();

  // ---- layer 2 ----
  {
    v8f c = gemm_tile(bufQ, 256, 256, w2, wv * 16, lane);
    if (lane < 16) {
      int nf = wv * 16 + lane;
      #pragma unroll
      for (int v = 0; v < 4; ++v) bufP[v * 256 + nf] = c[v];
    }
  }
  __syncthreads();
  if (tid < 256) {
    float y0 = bufP[0 * 256 + tid] + b2[tid];
    float y1 = bufP[1 * 256 + tid] + b2[tid];
    float y2 = bufP[2 * 256 + tid] + b2[tid];
    float y3 = bufP[3 * 256 + tid] + b2[tid];
    float mu = 0.25f * (y0 + y1 + y2 + y3);
    float d0 = y0 - mu, d1 = y1 - mu, d2 = y2 - mu, d3 = y3 - mu;
    float var = 0.25f * (d0 * d0 + d1 * d1 + d2 * d2 + d3 * d3);
    float sc = g2[tid] * rsqrtf(var + 1e-5f);
    float sh = be2[tid];
    bufQ[0 * 256 + tid] = fmaxf(sc * d0 + sh, 0.0f);
    bufQ[1 * 256 + tid] = fmaxf(sc * d1 + sh, 0.0f);
    bufQ[2 * 256 + tid] = fmaxf(sc * d2 + sh, 0.0f);
    bufQ[3 * 256 + tid] = fmaxf(sc * d3 + sh, 0.0f);
  }
  __syncthreads();

  // ---- layer 3 ----
  {
    v8f c = gemm_tile(bufQ, 256, 256, w3, wv * 16, lane);
    if (lane < 16) {
      int nf = wv * 16 + lane;
      #pragma unroll
      for (int v = 0; v < 4; ++v) bufP[v * 256 + nf] = c[v];
    }
  }
  __syncthreads();
  if (tid < 256) {
    float y0 = bufP[0 * 256 + tid] + b3[tid];
    float y1 = bufP[1 * 256 + tid] + b3[tid];
    float y2 = bufP[2 * 256 + tid] + b3[tid];
    float y3 = bufP[3 * 256 + tid] + b3[tid];
    float mu = 0.25f * (y0 + y1 + y2 + y3);
    float d0 = y0 - mu, d1 = y1 - mu, d2 = y2 - mu, d3 = y3 - mu;
    float var = 0.25f * (d0 * d0 + d1 * d1 + d2 * d2 + d3 * d3);
    float sc = g3[tid] * rsqrtf(var + 1e-5f);
    float sh = be3[tid];
    bufQ[0 * 256 + tid] = fmaxf(sc * d0 + sh, 0.0f);
    bufQ[1 * 256 + tid] = fmaxf(sc * d1 + sh, 0.0f);
    bufQ[2 * 256 + tid] = fmaxf(sc * d2 + sh, 0.0f);
    bufQ[3 * 256 + tid] = fmaxf(sc * d3 + sh, 0.0f);
  }
  __syncthreads();

  // ---- layer 4: [4,256] @ [256,3072]^T -> q, generated, Q ws ----
  for (int it = 0; it < 12; ++it) {
    int nf0 = (wv + 16 * it) * 16;
    v8f c = gemm_tile(bufQ, 256, 256, w4, nf0, lane);
    if (lane < 16) {
      int nf = nf0 + lane;
      float bias = b4[nf];
      int ci = nf >> 10;           // coordinate channel 0..2
      int mm = nf & 1023;          // query point index
      #pragma unroll
      for (int v = 0; v < 4; ++v) {
        float val = c[v] + bias;
        gen_out[((v << 10) + mm) * 3 + ci] = val;      // generated [B,M,3]
        Qws[(((v << 10) + mm) << 2) + ci] = val;       // Q float4 (w unused)
      }
    }
  }
}

// ---------------------------------------------------------------------------
// Score one 512-column chunk into an LDS staging buffer (this wave's 2 tiles).
// Also prefetches the chunk two steps ahead (global_prefetch_b8).
// ---------------------------------------------------------------------------
__device__ __forceinline__ void score_chunk(const float* __restrict__ Pf, int b,
                                            int cbase, v2f a, int lane, int wv,
                                            float* dst) {
  #pragma unroll
  for (int tt = 0; tt < 2; ++tt) {
    int t = 2 * wv + tt;
    int n = cbase + t * 16 + (lane & 15);
    const float* src = Pf + (((size_t)(b * NPTS + n)) << 2) +
                       ((lane < 16) ? 0 : 2);
    v2f bv = *(const v2f*)src;
    __builtin_prefetch(src + 2 * CHUNK * 4, 0, 1);   // warm chunk ch+2
    v8f c = {};
    c = __builtin_amdgcn_wmma_f32_16x16x4_f32(false, a, false, bv, (short)0, c,
                                              false, false);
    int base = (t * 32 + lane) * 8;
    *(float4*)&dst[base]     = make_float4(c[0], c[1], c[2], c[3]);
    *(float4*)&dst[base + 4] = make_float4(c[4], c[5], c[6], c[7]);
  }
}

// ---------------------------------------------------------------------------
// Kernel C: WMMA scoring (s = |x|^2 - 2 q.x), per-row top-16, softmax combine.
// One workgroup = one (b, 16-row m-tile); 16 waves; wave w selects row w.
// Double-buffered staging: one barrier per chunk, scoring of chunk ch+1
// overlaps selection of chunk ch.
// ---------------------------------------------------------------------------
__global__ __launch_bounds__(512) void knn_project_kernel(
    const float4* __restrict__ P, const float4* __restrict__ Q,
    const float* __restrict__ temp, float* __restrict__ out) {
  __shared__ float sc[2 * BUFSZ];         // 64 KB double-buffered staging
  unsigned int* topn = (unsigned int*)sc; // aliased after final selection

  int b = blockIdx.x >> 6;                // 64 m-tiles per batch
  int mt = blockIdx.x & 63;
  int tid = threadIdx.x;
  int lane = tid & 31;
  int wv = tid >> 5;                      // wave id == selected row
  int bm0 = b * MQ + mt * 16;

  // A-matrix registers: row m = lane&15, K split by lane half.
  int mA = lane & 15;
  float4 qa = Q[bm0 + mA];
  v2f a;
  if (lane < 16) { a.x = -2.0f * qa.x; a.y = -2.0f * qa.y; }
  else           { a.x = -2.0f * qa.z; a.y = 1.0f; }

  // per-lane running top-16 (64-bit keys: sortable(score)<<32 | index)
  unsigned long long arr[16];
  #pragma unroll
  for (int i = 0; i < 16; ++i) arr[i] = ~0ULL;
  unsigned long long worst = ~0ULL;
  int wslot = 0;

  int j = lane & 15;
  int half = lane >> 4;
  int rl = j + ((wv & 8) ? 16 : 0);       // source lane of row wv, col j
  int rv = wv & 7;                        // source VGPR slot of row wv
  const float* Pf = (const float*)P;

  const int NCH = NPTS / CHUNK;           // 64 chunks
  score_chunk(Pf, b, 0, a, lane, wv, sc); // prologue: chunk 0 -> buffer 0
  __syncthreads();

  for (int ch = 0; ch < NCH; ++ch) {
    // score next chunk into the other buffer (overlaps with selection below)
    if (ch + 1 < NCH)
      score_chunk(Pf, b, (ch + 1) * CHUNK, a, lane, wv,
                  sc + (((ch + 1) & 1) ? BUFSZ : 0));
    // selection for row wv from current buffer
    const float* scr = sc + ((ch & 1) ? BUFSZ : 0);
    int cbase = ch * CHUNK;
    #pragma unroll
    for (int i = 0; i < 16; ++i) {
      int t = half + 2 * i;
      float s = scr[(t * 32 + rl) * 8 + rv];
      unsigned int su = __float_as_uint(s);
      su = (su & 0x80000000u) ? ~su : (su | 0x80000000u);
      unsigned int n = (unsigned int)(cbase + t * 16 + j);
      unsigned long long key = ((unsigned long long)su << 32) | n;
      if (key < worst) {
        arr[wslot] = key;
        worst = arr[0]; wslot = 0;
        #pragma unroll
        for (int z = 1; z < 16; ++z)
          if (arr[z] > worst) { worst = arr[z]; wslot = z; }
      }
    }
    __syncthreads();
  }

  // --- wave-wide extraction of the 16 smallest keys (ascending) ---
  #pragma unroll 1
  for (int r = 0; r < 16; ++r) {
    unsigned long long mn = arr[0];
    #pragma unroll
    for (int z = 1; z < 16; ++z) mn = (arr[z] < mn) ? arr[z] : mn;
    #pragma unroll
    for (int off = 16; off > 0; off >>= 1) {
      unsigned int lo = (unsigned int)mn;
      unsigned int hi = (unsigned int)(mn >> 32);
      lo = __shfl_xor(lo, off, 32);
      hi = __shfl_xor(hi, off, 32);
      unsigned long long o = ((unsigned long long)hi << 32) | lo;
      mn = (o < mn) ? o : mn;
    }
    #pragma unroll
    for (int z = 0; z < 16; ++z)
      if (arr[z] == mn) arr[z] = ~0ULL;    // keys unique -> one owner
    if (lane == 0) topn[wv * 16 + r] = (unsigned int)mn;
  }
  __syncthreads();

  // --- soft projection: softmax over K=16 neighbors, weighted combine ---
  float t0 = temp[0];
  float sigma = fmaxf(t0 * t0, 0.01f);
  float4 qq = Q[bm0 + wv];
  int k = lane & 15;                       // both lane halves duplicate work
  unsigned int n = topn[wv * 16 + k];
  float4 g = P[b * NPTS + n];
  float dx = g.x - qq.x, dy = g.y - qq.y, dz = g.z - qq.z;
  float d = (dx * dx + dy * dy + dz * dz) / sigma;
  float mind = d;
  #pragma unroll
  for (int off = 8; off > 0; off >>= 1) mind = fminf(mind, __shfl_xor(mind, off, 32));
  float e = __expf(mind - d);
  float se = e;
  #pragma unroll
  for (int off = 8; off > 0; off >>= 1) se += __shfl_xor(se, off, 32);
  float wk = e / se;
  float px = wk * g.x, py = wk * g.y, pz = wk * g.z;
  #pragma unroll
  for (int off = 8; off > 0; off >>= 1) {
    px += __shfl_xor(px, off, 32);
    py += __shfl_xor(py, off, 32);
    pz += __shfl_xor(pz, off, 32);
  }
  if (lane == 0) {
    int ob = NB * MQ * 3 + (bm0 + wv) * 3;   // sampled follows generated
    out[ob + 0] = px;
    out[ob + 1] = py;
    out[ob + 2] = pz;
  }
}

// ---------------------------------------------------------------------------
extern "C" void kernel_launch(void* const* d_in, const int* in_sizes, int n_in,
                              void* d_out, int out_size, void* d_ws, size_t ws_size,
                              hipStream_t stream) {
  const float* x    = (const float*)d_in[0];
  const float* gf   = (const float*)d_in[1];
  const float* w1   = (const float*)d_in[2];
  const float* b1   = (const float*)d_in[3];
  const float* w2   = (const float*)d_in[4];
  const float* b2   = (const float*)d_in[5];
  const float* w3   = (const float*)d_in[6];
  const float* b3   = (const float*)d_in[7];
  const float* w4   = (const float*)d_in[8];
  const float* b4   = (const float*)d_in[9];
  const float* g1   = (const float*)d_in[10];
  const float* be1  = (const float*)d_in[11];
  const float* g2   = (const float*)d_in[12];
  const float* be2  = (const float*)d_in[13];
  const float* g3   = (const float*)d_in[14];
  const float* be3  = (const float*)d_in[15];
  const float* temp = (const float*)d_in[16];

  float4* P = (float4*)d_ws;                                   // B*N float4
  float4* Q = (float4*)((char*)d_ws + (size_t)NB * NPTS * 16); // B*M float4
  float* out = (float*)d_out;

  pack_x_kernel<<<(NB * NPTS) / 256, 256, 0, stream>>>(x, P);
  mlp_q_kernel<<<1, 512, 0, stream>>>(gf, w1, b1, w2, b2, w3, b3, w4, b4,
                                      g1, be1, g2, be2, g3, be3, out, (float*)Q);
  knn_project_kernel<<<NB * (MQ / 16), 512, 0, stream>>>(P, Q, temp, out);
}